// graphNetwork_nodesOnly_18451179503912
// MI455X (gfx1250) — compile-verified
//
#include <hip/hip_runtime.h>
#include <hip/hip_bf16.h>
#include <math.h>

// ---------------------------------------------------------------------------
// CDNA5 (gfx1250, wave32) implementation of the graph network reference.
// Strategy: never materialize the 4096x4096 affinity matrix D. Compute its
// mean/var statistics on the fly from WMMA-tiled Gram blocks (f^T f), then
// recompute affinities only at the 69,632 edge positions. All GEMMs run on
// v_wmma_f32_16x16x4_f32 with 2x2 register blocking (32x32 output per wave):
// 8 dword loads feed 4 WMMAs -> 32 FLOP/dword, double the density of a
// single-tile scheme.
// ---------------------------------------------------------------------------

typedef float v2f __attribute__((ext_vector_type(2)));
typedef float v8f __attribute__((ext_vector_type(8)));

__device__ __forceinline__ float wred_sum(float v) {
#pragma unroll
  for (int o = 16; o > 0; o >>= 1) v += __shfl_xor(v, o, 32);
  return v;
}
__device__ __forceinline__ float wred_max(float v) {
#pragma unroll
  for (int o = 16; o > 0; o >>= 1) v = fmaxf(v, __shfl_xor(v, o, 32));
  return v;
}

// ---------------------------------------------------------------------------
__global__ void k_zero(float* __restrict__ p, int n) {
  int i = blockIdx.x * blockDim.x + threadIdx.x;
  if (i < n) p[i] = 0.f;
}

// concat[0:n) = src, concat[n:2n) = 0  (xn copy + Laplacian zero)
__global__ void k_copy_zero(float* __restrict__ dst, const float* __restrict__ src, int n) {
  int i = blockIdx.x * blockDim.x + threadIdx.x;
  if (i < n) { dst[i] = src[i]; dst[n + i] = 0.f; }
}

// ---------------------------------------------------------------------------
// Generic C-major GEMM: out[m][n] = sum_k A[m][k] * B[k][n]
// One 32x32 output block per wave = 2x2 grid of 16x16 WMMA fragments.
// Epilogue: optional bias[m], optional out = addend + alpha*acc, optional ELU,
// optional transposed store out[n*M + m].
// Fragment layouts (ISA 7.12.2, f32):
//   A 16x4 : lane lm = M row, vgpr v: K = v + 2*half
//   B 4x16 : lane lm = N col, vgpr v: K = v + 2*half
//   C 16x16: lane lm = N col, vgpr v: M = v + 8*half
// ---------------------------------------------------------------------------
__global__ void k_gemm_wmma(const float* __restrict__ A, const float* __restrict__ B,
                            float* __restrict__ out, const float* __restrict__ bias,
                            const float* __restrict__ addend, float alpha,
                            int M, int K, int N, int act, int transStore) {
  const int lane = threadIdx.x & 31;
  const int wid  = threadIdx.x >> 5;
  const int gw   = blockIdx.x * (blockDim.x >> 5) + wid;
  const int nblk = N >> 5;
  const int mb = gw / nblk, nb = gw % nblk;
  if (mb >= (M >> 5)) return;                 // wave-uniform exit (EXEC stays full)
  const int m0 = mb << 5, n0 = nb << 5;
  const int half = lane >> 4, lm = lane & 15;

  v8f acc[2][2];
#pragma unroll
  for (int i = 0; i < 2; ++i)
#pragma unroll
    for (int j = 0; j < 2; ++j) acc[i][j] = (v8f){0.f,0.f,0.f,0.f,0.f,0.f,0.f,0.f};

  const float* arow0 = A + (size_t)(m0 + lm) * K;
  const float* arow1 = arow0 + (size_t)16 * K;
  for (int kk = 0; kk < K; kk += 4) {
    const int ka = kk + 2 * half;
    v2f a0; a0.x = arow0[ka];  a0.y = arow0[ka + 1];
    v2f a1; a1.x = arow1[ka];  a1.y = arow1[ka + 1];
    const float* bp0 = B + (size_t)ka * N + n0;
    const float* bp1 = B + (size_t)(ka + 1) * N + n0;
    v2f b0; b0.x = bp0[lm];      b0.y = bp1[lm];
    v2f b1; b1.x = bp0[16 + lm]; b1.y = bp1[16 + lm];
    acc[0][0] = __builtin_amdgcn_wmma_f32_16x16x4_f32(false, a0, false, b0, (short)0, acc[0][0], false, false);
    acc[0][1] = __builtin_amdgcn_wmma_f32_16x16x4_f32(false, a0, false, b1, (short)0, acc[0][1], false, false);
    acc[1][0] = __builtin_amdgcn_wmma_f32_16x16x4_f32(false, a1, false, b0, (short)0, acc[1][0], false, false);
    acc[1][1] = __builtin_amdgcn_wmma_f32_16x16x4_f32(false, a1, false, b1, (short)0, acc[1][1], false, false);
  }

#pragma unroll
  for (int ai = 0; ai < 2; ++ai)
#pragma unroll
    for (int bj = 0; bj < 2; ++bj) {
#pragma unroll
      for (int v = 0; v < 8; ++v) {
        const int m = m0 + 16 * ai + v + 8 * half;
        const int n = n0 + 16 * bj + lm;
        float val = acc[ai][bj][v];
        if (bias)   val += bias[m];
        if (addend) val = addend[(size_t)m * N + n] + alpha * val;
        if (act == 1) val = (val > 0.f) ? val : (expf(val) - 1.f);   // ELU
        if (transStore) out[(size_t)n * M + m] = val;
        else            out[(size_t)m * N + n] = val;
      }
    }
}

// ---------------------------------------------------------------------------
// sq[n] = sum_c f[c][n]^2
__global__ void k_colsq(const float* __restrict__ f, float* __restrict__ sq, int C, int N) {
  int n = blockIdx.x * blockDim.x + threadIdx.x;
  if (n >= N) return;
  float s = 0.f;
  for (int c = 0; c < C; ++c) { float v = f[(size_t)c * N + n]; s = fmaf(v, v, s); }
  sq[n] = s;
}

// ---------------------------------------------------------------------------
// Gram-block statistics: one 32x32 block of G = f^T f per wave (2x2 WMMA
// fragments), then accumulate sum / sumsq of relu(sq_i + sq_j - 2*G_ij) into
// stats[0..1]. Symmetry at 32-block granularity: only bj >= bi blocks;
// off-diagonal blocks weighted x2 (diagonal blocks already contain both
// mirrored triangles, so x1 is exact).
// ---------------------------------------------------------------------------
__global__ void k_gram_stats(const float* __restrict__ f, const float* __restrict__ sq,
                             float* __restrict__ stats, int N, int C) {
  const int lane = threadIdx.x & 31;
  const int wid  = threadIdx.x >> 5;
  const int bj   = blockIdx.x * (blockDim.x >> 5) + wid;   // 32-block column
  const int bi   = blockIdx.y;                              // 32-block row
  const int nblk = N >> 5;
  if (bj >= nblk || bj < bi) return;          // wave-uniform
  const int i0 = bi << 5, j0 = bj << 5;
  const int half = lane >> 4, lm = lane & 15;

  v8f acc[2][2];
#pragma unroll
  for (int i = 0; i < 2; ++i)
#pragma unroll
    for (int j = 0; j < 2; ++j) acc[i][j] = (v8f){0.f,0.f,0.f,0.f,0.f,0.f,0.f,0.f};

  for (int kk = 0; kk < C; kk += 4) {
    const int ka = kk + 2 * half;
    const float* r0 = f + (size_t)ka * N;
    const float* r1 = f + (size_t)(ka + 1) * N;
    v2f a0; a0.x = r0[i0 + lm];      a0.y = r1[i0 + lm];
    v2f a1; a1.x = r0[i0 + 16 + lm]; a1.y = r1[i0 + 16 + lm];
    v2f b0; b0.x = r0[j0 + lm];      b0.y = r1[j0 + lm];
    v2f b1; b1.x = r0[j0 + 16 + lm]; b1.y = r1[j0 + 16 + lm];
    acc[0][0] = __builtin_amdgcn_wmma_f32_16x16x4_f32(false, a0, false, b0, (short)0, acc[0][0], false, false);
    acc[0][1] = __builtin_amdgcn_wmma_f32_16x16x4_f32(false, a0, false, b1, (short)0, acc[0][1], false, false);
    acc[1][0] = __builtin_amdgcn_wmma_f32_16x16x4_f32(false, a1, false, b0, (short)0, acc[1][0], false, false);
    acc[1][1] = __builtin_amdgcn_wmma_f32_16x16x4_f32(false, a1, false, b1, (short)0, acc[1][1], false, false);
  }

  const float wgt = (bi == bj) ? 1.f : 2.f;
  const float sqj0 = sq[j0 + lm];
  const float sqj1 = sq[j0 + 16 + lm];
  float s1 = 0.f, s2 = 0.f;
#pragma unroll
  for (int ai = 0; ai < 2; ++ai) {
#pragma unroll
    for (int v = 0; v < 8; ++v) {
      const int i = i0 + 16 * ai + v + 8 * half;
      const float sqi = sq[i];
      float r0 = fmaxf(sqi + sqj0 - 2.f * acc[ai][0][v], 0.f);
      float r1 = fmaxf(sqi + sqj1 - 2.f * acc[ai][1][v], 0.f);
      s1 += r0 + r1;
      s2 = fmaf(r0, r0, s2);
      s2 = fmaf(r1, r1, s2);
    }
  }
  s1 = wred_sum(s1 * wgt);
  s2 = wred_sum(s2 * wgt);
  if (lane == 0) { atomicAdd(stats + 0, s1); atomicAdd(stats + 1, s2); }
}

// stats[2] = -2 / std(D, ddof=1)
__global__ void k_final_dstats(float* __restrict__ stats, double nelem) {
  if (threadIdx.x == 0 && blockIdx.x == 0) {
    double s1 = (double)stats[0], s2 = (double)stats[1];
    double var = (s2 - s1 * s1 / nelem) / (nelem - 1.0);
    double sd  = sqrt(var > 0.0 ? var : 0.0);
    stats[2] = (float)(-2.0 / sd);
  }
}

// ---------------------------------------------------------------------------
// Per-edge affinity: w[e] = exp(scale * relu(sq_i + sq_j - 2 f_i . f_j));
// accumulate deg[j] += w.  One wave per edge; lanes split the C dot product.
// Self-loops (e >= E0) use i = j = e - E0 -> w = exp(0) = 1.
// ---------------------------------------------------------------------------
__global__ void k_edge_w(const int* __restrict__ ei, const int* __restrict__ ej,
                         const float* __restrict__ f, const float* __restrict__ sq,
                         const float* __restrict__ stats,
                         float* __restrict__ w, float* __restrict__ deg,
                         int E0, int N, int C) {
  const int lane = threadIdx.x & 31;
  const int e = blockIdx.x * (blockDim.x >> 5) + (threadIdx.x >> 5);
  const int E = E0 + N;
  if (e >= E) return;
  const int i = (e < E0) ? ei[e] : (e - E0);
  const int j = (e < E0) ? ej[e] : (e - E0);
  float dot = 0.f;
  for (int c = lane; c < C; c += 32)
    dot = fmaf(f[(size_t)c * N + i], f[(size_t)c * N + j], dot);
  dot = wred_sum(dot);
  if (lane == 0) {
    float r  = fmaxf(sq[i] + sq[j] - 2.f * dot, 0.f);
    float wv = expf(stats[2] * r);
    w[e] = wv;
    atomicAdd(deg + j, wv);
  }
}

// W[e] = dis[i] * w[e] * dis[j],  dis = deg>0 ? rsqrt(deg) : 0  (in place)
__global__ void k_edge_norm(const int* __restrict__ ei, const int* __restrict__ ej,
                            const float* __restrict__ deg, float* __restrict__ w,
                            int E0, int N) {
  int e = blockIdx.x * blockDim.x + threadIdx.x;
  const int E = E0 + N;
  if (e >= E) return;
  const int i = (e < E0) ? ei[e] : (e - E0);
  const int j = (e < E0) ? ej[e] : (e - E0);
  const float di = deg[i] > 0.f ? rsqrtf(deg[i]) : 0.f;
  const float dj = deg[j] > 0.f ? rsqrtf(deg[j]) : 0.f;
  w[e] = di * w[e] * dj;
}

// ---------------------------------------------------------------------------
// Laplacian scatter: lap[c][i] += W^2 * (x[c][i]-x[c][j]); lap[c][j] -= same.
// One wave per edge, lanes over channels.
// ---------------------------------------------------------------------------
__global__ void k_lap_scatter(const int* __restrict__ ei, const int* __restrict__ ej,
                              const float* __restrict__ W, const float* __restrict__ x,
                              float* __restrict__ lap, int E0, int N, int C) {
  const int lane = threadIdx.x & 31;
  const int e = blockIdx.x * (blockDim.x >> 5) + (threadIdx.x >> 5);
  const int E = E0 + N;
  if (e >= E) return;
  const int i = (e < E0) ? ei[e] : (e - E0);
  const int j = (e < E0) ? ej[e] : (e - E0);
  if (i == j) return;                          // self-loop contributes exactly 0
  const float Wv = W[e];
  const float coef = Wv * Wv;
  for (int c = lane; c < C; c += 32) {
    const float d = coef * (x[(size_t)c * N + i] - x[(size_t)c * N + j]);
    atomicAdd(&lap[(size_t)c * N + i],  d);
    atomicAdd(&lap[(size_t)c * N + j], -d);
  }
}

// ---------------------------------------------------------------------------
// Whole-tensor layernorm helpers
__global__ void k_reduce_sums(const float* __restrict__ x, float* __restrict__ stats, int n) {
  __shared__ float s1s[8], s2s[8];
  float s1 = 0.f, s2 = 0.f;
  for (int i = blockIdx.x * blockDim.x + threadIdx.x; i < n; i += gridDim.x * blockDim.x) {
    const float v = x[i];
    s1 += v;
    s2 = fmaf(v, v, s2);
  }
  s1 = wred_sum(s1); s2 = wred_sum(s2);
  const int lane = threadIdx.x & 31, wid = threadIdx.x >> 5, nw = blockDim.x >> 5;
  if (lane == 0) { s1s[wid] = s1; s2s[wid] = s2; }
  __syncthreads();
  if (wid == 0) {
    s1 = (lane < nw) ? s1s[lane] : 0.f;
    s2 = (lane < nw) ? s2s[lane] : 0.f;
    s1 = wred_sum(s1); s2 = wred_sum(s2);
    if (lane == 0) { atomicAdd(stats + 0, s1); atomicAdd(stats + 1, s2); }
  }
}

__global__ void k_final_ln(float* __restrict__ stats, float n) {
  if (threadIdx.x == 0 && blockIdx.x == 0) {
    const float m   = stats[0] / n;
    const float var = stats[1] / n - m * m;
    stats[2] = m;
    stats[3] = rsqrtf(var + 1e-5f);
  }
}

__global__ void k_tanh_norm(float* __restrict__ x, const float* __restrict__ stats, int n) {
  int i = blockIdx.x * blockDim.x + threadIdx.x;
  if (i < n) x[i] = tanhf((x[i] - stats[2]) * stats[3]);
}

// ---------------------------------------------------------------------------
// In-place row-wise log_softmax over K columns; one workgroup per row.
__global__ void k_log_softmax(float* __restrict__ x, int K) {
  __shared__ float red[8];
  __shared__ float s_mx, s_ls;
  float* row = x + (size_t)blockIdx.x * K;
  const int t = threadIdx.x, lane = t & 31, wid = t >> 5, nw = blockDim.x >> 5;

  float mx = -3.402823466e38f;
  for (int k = t; k < K; k += blockDim.x) mx = fmaxf(mx, row[k]);
  mx = wred_max(mx);
  if (lane == 0) red[wid] = mx;
  __syncthreads();
  if (t == 0) {
    float m = red[0];
    for (int i = 1; i < nw; ++i) m = fmaxf(m, red[i]);
    s_mx = m;
  }
  __syncthreads();
  mx = s_mx;

  float sum = 0.f;
  for (int k = t; k < K; k += blockDim.x) sum += expf(row[k] - mx);
  sum = wred_sum(sum);
  __syncthreads();
  if (lane == 0) red[wid] = sum;
  __syncthreads();
  if (t == 0) {
    float s = 0.f;
    for (int i = 0; i < nw; ++i) s += red[i];
    s_ls = logf(s);
  }
  __syncthreads();
  const float ls = s_ls;
  for (int k = t; k < K; k += blockDim.x) row[k] = row[k] - mx - ls;
}

// ---------------------------------------------------------------------------
extern "C" void kernel_launch(void* const* d_in, const int* in_sizes, int n_in,
                              void* d_out, int out_size, void* d_ws, size_t ws_size,
                              hipStream_t stream) {
  (void)n_in; (void)out_size; (void)ws_size;
  const float* xn0 = (const float*)d_in[0];   // [1, 64, 4096]
  const float* K1  = (const float*)d_in[1];   // [256, 64]
  const float* K2  = (const float*)d_in[2];   // [256, 256]
  const float* KNc = (const float*)d_in[3];   // [256, 256]
  const float* KN1 = (const float*)d_in[4];   // [8, 256, 512]
  const float* KN2 = (const float*)d_in[5];   // [8, 256, 256]
  const float* l1w = (const float*)d_in[6];   // [256, 256]
  const float* l1b = (const float*)d_in[7];   // [256]
  const float* l2w = (const float*)d_in[8];   // [1024, 256]
  const float* l2b = (const float*)d_in[9];   // [1024]
  const int*   ei  = (const int*)d_in[10];    // [E0]
  const int*   ej  = (const int*)d_in[11];    // [E0]

  const int N = 4096, CIN = 64, C = 256, NLAYER = 8, OUTK = 1024;
  const int E0 = in_sizes[10];
  const int E  = E0 + N;

  // workspace layout (floats)
  float* xn      = (float*)d_ws;            // [C][N]
  float* cat     = xn  + (size_t)C * N;     // [2C][N] (rows 0..C-1: xn copy; C..: lap)
  float* hid     = cat + (size_t)2 * C * N; // [C][N]
  float* wbuf    = hid + (size_t)C * N;     // [E]
  float* sq      = wbuf + E;                // [N]
  float* deg     = sq + N;                  // [N]
  float* stats   = deg + N;                 // [0..2] D-stats, [4..7] LN-stats
  float* statsLN = stats + 4;
  float* lap     = cat + (size_t)C * N;

  auto cdiv = [](int a, int b) { return (a + b - 1) / b; };

  auto gemm = [&](const float* A, const float* B, float* out, const float* bias,
                  const float* addend, float alpha, int M, int K, int act, int trans) {
    const int tiles = (M >> 5) * (N >> 5);   // 32x32 blocks, one per wave
    k_gemm_wmma<<<cdiv(tiles, 8), 256, 0, stream>>>(A, B, out, bias, addend, alpha,
                                                    M, K, N, act, trans);
  };

  auto layernorm_tanh = [&](float* x, int n) {
    k_zero<<<1, 64, 0, stream>>>(statsLN, 2);
    k_reduce_sums<<<1024, 256, 0, stream>>>(x, statsLN, n);
    k_final_ln<<<1, 32, 0, stream>>>(statsLN, (float)n);
    k_tanh_norm<<<cdiv(n, 256), 256, 0, stream>>>(x, statsLN, n);
  };

  // ---- opening double_layer: hid = K1 @ xn0; LN+tanh; xn = K2 @ hid ----
  gemm(K1, xn0, hid, nullptr, nullptr, 0.f, C, CIN, 0, 0);
  layernorm_tanh(hid, C * N);
  gemm(K2, hid, xn, nullptr, nullptr, 0.f, C, C, 0, 0);

  // ---- 8 heat-propagation layers ----
  for (int it = 0; it < NLAYER; ++it) {
    // affinity statistics (no D materialization)
    k_zero<<<1, 64, 0, stream>>>(stats, 3);
    k_zero<<<cdiv(N, 256), 256, 0, stream>>>(deg, N);
    k_colsq<<<cdiv(N, 256), 256, 0, stream>>>(xn, sq, C, N);
    {
      const int nblk = N >> 5;                 // 128 32-blocks per dim
      dim3 g(cdiv(nblk, 8), nblk);             // early-exit below diagonal
      k_gram_stats<<<g, 256, 0, stream>>>(xn, sq, stats, N, C);
    }
    k_final_dstats<<<1, 32, 0, stream>>>(stats, (double)N * (double)N);

    // edge weights, degree normalization
    k_edge_w<<<cdiv(E, 8), 256, 0, stream>>>(ei, ej, xn, sq, stats, wbuf, deg, E0, N, C);
    k_edge_norm<<<cdiv(E, 256), 256, 0, stream>>>(ei, ej, deg, wbuf, E0, N);

    // varlet features: cat = [xn ; lap], lap via scatter
    k_copy_zero<<<cdiv(C * N, 256), 256, 0, stream>>>(cat, xn, C * N);
    k_lap_scatter<<<cdiv(E, 8), 256, 0, stream>>>(ei, ej, wbuf, xn, lap, E0, N, C);

    // double_layer + heat update (residual fused into second GEMM)
    gemm(KN1 + (size_t)it * C * (2 * C), cat, hid, nullptr, nullptr, 0.f, C, 2 * C, 0, 0);
    layernorm_tanh(hid, C * N);
    gemm(KN2 + (size_t)it * C * C, hid, xn, nullptr, xn, 0.1f, C, C, 0, 0);
  }

  // ---- close + MLP head ----
  gemm(KNc, xn, hid, nullptr, nullptr, 0.f, C, C, 0, 0);            // xc  [C][N]
  gemm(l1w, hid, cat, l1b, nullptr, 0.f, C, C, 1, 0);               // ELU [C][N]
  gemm(l2w, cat, (float*)d_out, l2b, nullptr, 0.f, OUTK, C, 0, 1);  // out [N][1024]
  k_log_softmax<<<N, 256, 0, stream>>>((float*)d_out, OUTK);
}